// SWAttention_AV_53085795778936
// MI455X (gfx1250) — compile-verified
//
#include <hip/hip_runtime.h>

typedef float v2f __attribute__((ext_vector_type(2)));
typedef float v8f __attribute__((ext_vector_type(8)));

#define B_    8
#define H_    8
#define N_    3136
#define D_    32
#define K2_   9
#define TILE_N 64          // rows per workgroup (4 waves x 16-row WMMA tiles)
#define AS_STRIDE 17       // padded attn_s row stride (avoids lo/hi lane bank clash)

__global__ __launch_bounds__(128) void swattn_av_kernel(
    const float* __restrict__ q,          // [B,H,N,32]
    const float* __restrict__ attn_local, // [B,H,N,9]
    const float* __restrict__ v,          // [B,H,N,32,9]
    const float* __restrict__ lt,         // [H,32,9]
    const float* __restrict__ bias,       // [H,N,9]
    float* __restrict__ out)              // [B,H,N,32]
{
    __shared__ float lt_s[D_ * 16];             // [k=0..31][n=0..15], zero-padded cols 9..15
    __shared__ float attn_s[TILE_N * AS_STRIDE];

    const int t    = threadIdx.x;
    const int wave = t >> 5;
    const int lane = t & 31;

    const int tiles_per_bh = N_ / TILE_N;       // 49
    const int bh = blockIdx.x / tiles_per_bh;
    const int nb = (blockIdx.x % tiles_per_bh) * TILE_N;
    const int h  = bh % H_;

    // ---- Phase 0: stage zero-padded learnable_tokens[h] (32x9 -> 32x16) in LDS ----
    {
        const float* ltb = lt + (size_t)h * (D_ * K2_);
        #pragma unroll
        for (int i = t; i < D_ * 16; i += 128) {
            const int row = i >> 4;             // k index 0..31
            const int col = i & 15;             // n index 0..15
            lt_s[i] = (col < K2_) ? ltb[row * K2_ + col] : 0.f;
        }
    }
    __syncthreads();

    // ---- Phase 1: attn[16x16] = q[16x32] @ lt[32x16] via WMMA f32 16x16x4 ----
    const int  nb2 = nb + wave * 16;            // this wave's 16-row sub-tile
    const int  r   = lane & 15;                 // A: row index / B: col index
    const bool hi  = lane >= 16;                // K-offset +2 half of the wave

    const float* qrow = q + ((size_t)bh * N_ + (nb2 + r)) * D_;

    v8f c = {0.f, 0.f, 0.f, 0.f, 0.f, 0.f, 0.f, 0.f};
    #pragma unroll
    for (int kk = 0; kk < 8; ++kk) {            // K = 32 as 8 steps of K4
        const int k0 = kk * 4 + (hi ? 2 : 0);
        v2f a, b;
        a.x = qrow[k0];
        a.y = qrow[k0 + 1];
        b.x = lt_s[(k0    ) * 16 + r];          // conflict-free, branchless
        b.y = lt_s[(k0 + 1) * 16 + r];
        c = __builtin_amdgcn_wmma_f32_16x16x4_f32(false, a, false, b,
                                                  (short)0, c, false, false);
    }

    // C layout: c[vv] = dot(q[row], lt[:,col]) with row = vv + (hi?8:0), col = r.
    // Store all 16 cols unconditionally (cols 9..15 are dead padding).
    #pragma unroll
    for (int vv = 0; vv < 8; ++vv) {
        const int row = vv + (hi ? 8 : 0);
        attn_s[(wave * 16 + row) * AS_STRIDE + r] = c[vv];
    }
    __syncthreads();

    // ---- Phase 1b: attn += bias + attn_local (coalesced cooperative pass) ----
    {
        const float* bsrc = bias       + ((size_t)h  * N_ + nb) * K2_;
        const float* asrc = attn_local + ((size_t)bh * N_ + nb) * K2_;
        #pragma unroll
        for (int i = t; i < TILE_N * K2_; i += 128) {
            const int nloc = i / K2_;
            const int k    = i - nloc * K2_;
            attn_s[nloc * AS_STRIDE + k] += bsrc[i] + asrc[i];
        }
    }
    __syncthreads();

    // ---- Phase 2: out[n,d] = sum_k attn[n,k] * v[n,d,k] (NT streaming) ----
    const int d = lane;                         // 0..31 == D
    #pragma unroll 4
    for (int rr = 0; rr < 16; ++rr) {
        const int nloc = wave * 16 + rr;
        const int n    = nb + nloc;
        const float* vrow = v + (((size_t)bh * N_ + n) * D_ + d) * K2_;
        if (rr < 12)                            // stay inside this tile's slab
            __builtin_prefetch(vrow + 4 * (D_ * K2_), 0, 0);
        float acc = 0.f;
        #pragma unroll
        for (int k = 0; k < K2_; ++k) {
            acc = fmaf(attn_s[nloc * AS_STRIDE + k],
                       __builtin_nontemporal_load(vrow + k), acc);
        }
        __builtin_nontemporal_store(acc, out + (((size_t)bh * N_ + n) * D_ + d));
    }
}

extern "C" void kernel_launch(void* const* d_in, const int* in_sizes, int n_in,
                              void* d_out, int out_size, void* d_ws, size_t ws_size,
                              hipStream_t stream) {
    (void)in_sizes; (void)n_in; (void)d_ws; (void)ws_size; (void)out_size;
    const float* q          = (const float*)d_in[0];
    const float* attn_local = (const float*)d_in[1];
    const float* v          = (const float*)d_in[2];
    const float* lt         = (const float*)d_in[3];
    const float* bias       = (const float*)d_in[4];
    float* out = (float*)d_out;

    const int grid = B_ * H_ * (N_ / TILE_N);   // 3136 workgroups
    swattn_av_kernel<<<grid, 128, 0, stream>>>(q, attn_local, v, lt, bias, out);
}